// transformer_block_4346506903633
// MI455X (gfx1250) — compile-verified
//
#include <hip/hip_runtime.h>
#include <math.h>

typedef __attribute__((ext_vector_type(2))) float v2f;
typedef __attribute__((ext_vector_type(8))) float v8f;

// Problem constants (from reference setup_inputs)
constexpr int CIN = 64, CO = 128, TT = 64, VV = 25, HH = 3, NB = 256;
constexpr int R  = NB * TT;          // 16384 rows (b,t)
constexpr int RB = 8;                // rows per workgroup
constexpr int MROWS = RB * VV;       // 200 flattened (r,j) rows per block
constexpr int MT1 = 13;              // ceil(200/16) M-tiles, rows 200..207 zero-padded

// LDS strides chosen for conflict-free WMMA fragment ds_loads
constexpr int XS_STRIDE = 68;        // Xs[m][c]   : (68*m)%64 = 4m -> lanes hit distinct banks
constexpr int WS_STRIDE = 136;       // Wsh[c][n]  : (2*136)%64 = 16 -> half-wave groups disjoint
constexpr int VS_STRIDE = 40;        // Vs[r][hj][o']: (2*40)%64 = 16
constexpr int VS_ROWS   = 76;        // (h,j)=75 padded to 76 (row 75 stays zero)
constexpr int A2_STRIDE = 84;        // A2s[i][hj] : (84*m)%64 = 20m, 16 distinct mod-4 banks

constexpr float EPSV = 1e-5f;

constexpr size_t Y_ELEMS  = (size_t)NB * CO * TT * VV;  // 52,428,800
constexpr size_t COEF_OFF = 512;                        // floats
constexpr size_t Y_OFF    = 1024;                       // floats (16B aligned)
constexpr size_t R_OFF    = 1024 + Y_ELEMS;             // floats (16B aligned)

// ---------------------------------------------------------------- K0: zero stats
__global__ void k0_zero_stats(float* __restrict__ ws) {
  int i = blockIdx.x * 256 + threadIdx.x;
  if (i < 1024) ws[i] = 0.f;
}

// ---------------------------------------------------------------- K1: fused two-stage WMMA
// Stage 1: [v_h | r] (n' x 128chunk) = X(n' x 64) @ W(64 x 128chunk)   (n' = (b,t,j))
// Stage 2: y_r(i x o) = PA_all(i x (h,j)) @ V_r((h,j) x o)  per row r, PA frags in regs
__global__ __launch_bounds__(256)
void gsa_fused_k1(const float* __restrict__ x,  const float* __restrict__ PA,
                  const float* __restrict__ Wv, const float* __restrict__ bv,
                  const float* __restrict__ Wr, const float* __restrict__ br,
                  float* __restrict__ ws)
{
  __shared__ float Xs [208 * XS_STRIDE];          // 56.6 KB  X block, rows 200..207 zero
  __shared__ float Wsh[CIN * WS_STRIDE];          // 34.8 KB  weight chunk (3 heads + residual)
  __shared__ float Vs [RB * VS_ROWS * VS_STRIDE]; // 97.3 KB  v (+bias) staged for stage 2
  __shared__ float A2s[32 * A2_STRIDE];           // 10.8 KB  PA_all, i padded to 32, hj to 76
  __shared__ float sY[CO], sY2[CO], sR[CO], sR2[CO];

  const int tid  = threadIdx.x;
  const int wave = tid >> 5;
  const int lane = tid & 31;
  const int lh   = lane >> 4;   // half-wave select (K/M split per ISA layouts)
  const int ll   = lane & 15;
  const int r0   = blockIdx.x * RB;

  for (int i = tid; i < 208 * XS_STRIDE; i += 256) Xs[i] = 0.f;
  for (int i = tid; i < RB * VS_ROWS * VS_STRIDE; i += 256) Vs[i] = 0.f;
  for (int i = tid; i < 32 * A2_STRIDE; i += 256) A2s[i] = 0.f;
  for (int i = tid; i < CO; i += 256) { sY[i] = 0.f; sY2[i] = 0.f; sR[i] = 0.f; sR2[i] = 0.f; }
  __syncthreads();

  // Load X block: Xs[m][c] = x[b, c, t, j], m = rl*25 + j (global coalesced over j)
  for (int idx = tid; idx < MROWS * CIN; idx += 256) {
    int c  = idx / MROWS;
    int m  = idx - c * MROWS;
    int rl = m / VV;
    int j  = m - rl * VV;
    int rg = r0 + rl;
    int b  = rg >> 6, t = rg & 63;
    Xs[m * XS_STRIDE + c] = x[(((size_t)(b * CIN + c)) * TT + t) * VV + j];
  }
  // Load PA into A2s[i][h*25+j]
  for (int p = tid; p < HH * VV * VV; p += 256) {
    int h = p / (VV * VV);
    int rem = p - h * VV * VV;
    int i = rem / VV, j = rem - i * VV;
    A2s[i * A2_STRIDE + h * VV + j] = PA[p];
  }
  __syncthreads();

  // Stage-2 A fragments (PA) are row-invariant: preload once per wave into regs.
  const int mi2 = (wave >> 1) & 1;   // i-tile (0: i=0..15, 1: i=16..31)
  const int no2 = wave & 1;          // o'-tile within 32-wide chunk
  v2f a2[19];
  #pragma unroll
  for (int ks = 0; ks < 19; ++ks) {
    int kk  = ks * 4 + 2 * lh;
    int row = mi2 * 16 + ll;
    a2[ks] = *(const v2f*)(A2s + row * A2_STRIDE + kk);
  }

  #pragma unroll 1
  for (int oc = 0; oc < 4; ++oc) {                // 4 chunks of 32 output channels
    __syncthreads();                              // prior stage-2 readers done
    // Load weight chunk: cols 0..95 = Wv heads, 96..127 = Wr (coalesced over c)
    for (int idx = tid; idx < CIN * 128; idx += 256) {
      int c = idx & 63;
      int ncol = idx >> 6;
      float w;
      if (ncol < 96) {
        int h = ncol >> 5, oo = ncol & 31;
        w = Wv[((size_t)(h * CO + oc * 32 + oo)) * CIN + c];
      } else {
        w = Wr[((size_t)(oc * 32 + (ncol - 96))) * CIN + c];
      }
      Wsh[c * WS_STRIDE + ncol] = w;
    }
    __syncthreads();

    // ---------------- Stage 1: each wave owns fixed column tile nt = wave
    v2f bf[16];
    #pragma unroll
    for (int ks = 0; ks < 16; ++ks) {             // preload B frags (held in regs)
      int kk = ks * 4 + 2 * lh;
      int col = wave * 16 + ll;
      bf[ks].x = Wsh[kk * WS_STRIDE + col];
      bf[ks].y = Wsh[(kk + 1) * WS_STRIDE + col];
    }
    const int ncol = wave * 16 + ll;              // wave<6 -> v cols, wave>=6 -> residual
    float bias; int h_c = 0, oo_c;
    if (wave < 6) { h_c = ncol >> 5; oo_c = ncol & 31; bias = bv[h_c * CO + oc * 32 + oo_c]; }
    else          { oo_c = ncol - 96;               bias = br[oc * 32 + oo_c]; }
    const int o_res = oc * 32 + oo_c;             // global channel for residual waves

    #pragma unroll 1
    for (int mt = 0; mt < MT1; ++mt) {
      v8f acc = {};
      #pragma unroll
      for (int ks = 0; ks < 16; ++ks) {           // K = 64 = 16 x (K=4)
        int kk = ks * 4 + 2 * lh;
        int row = mt * 16 + ll;
        v2f a = *(const v2f*)(Xs + row * XS_STRIDE + kk);
        acc = __builtin_amdgcn_wmma_f32_16x16x4_f32(false, a, false, bf[ks],
                                                    (short)0, acc, false, false);
      }
      // Epilogue: rows are full except mt==12, where validity is exactly lh==0.
      const bool rows_ok = (mt < 12) || (lh == 0);
      if (wave < 6) {                             // uniform: stage v+bias into Vs
        if (rows_ok) {
          #pragma unroll
          for (int g = 0; g < 8; ++g) {
            int m = mt * 16 + g + 8 * lh;         // D layout: VGPR g -> row g + 8*(lane>=16)
            int rl = m / VV, j = m - rl * VV;
            Vs[(rl * VS_ROWS + h_c * VV + j) * VS_STRIDE + oo_c] = acc[g] + bias;
          }
        }
      } else {                                    // uniform: residual -> global + stats
        float psum = 0.f, psum2 = 0.f;
        if (rows_ok) {
          #pragma unroll
          for (int g = 0; g < 8; ++g) {
            int m = mt * 16 + g + 8 * lh;
            float val = acc[g] + bias;
            int rl = m / VV, j = m - rl * VV;
            int rg = r0 + rl, b = rg >> 6, t = rg & 63;
            ws[R_OFF + (((size_t)(b * CO + o_res)) * TT + t) * VV + j] = val;
            psum += val; psum2 += val * val;
          }
        }
        atomicAdd(&sR[o_res], psum);
        atomicAdd(&sR2[o_res], psum2);
      }
    }
    __syncthreads();

    // ---------------- Stage 2: per-row y_r = PA_all @ V_r, A frags already in regs
    const int colo = no2 * 16 + ll;
    const int o2   = oc * 32 + colo;
    #pragma unroll 1
    for (int k = 0; k < 4; ++k) {
      int rl = (wave >> 2) + 2 * k;               // each wave: 4 of the 8 rows
      v8f acc = {};
      #pragma unroll
      for (int ks = 0; ks < 19; ++ks) {           // K = 75 -> 76 = 19 x (K=4)
        int kr = ks * 4 + 2 * lh;
        v2f b2;
        b2.x = Vs[(rl * VS_ROWS + kr)     * VS_STRIDE + colo];
        b2.y = Vs[(rl * VS_ROWS + kr + 1) * VS_STRIDE + colo];
        acc = __builtin_amdgcn_wmma_f32_16x16x4_f32(false, a2[ks], false, b2,
                                                    (short)0, acc, false, false);
      }
      int rg = r0 + rl, b = rg >> 6, t = rg & 63;
      const size_t ybase = Y_OFF + (((size_t)(b * CO + o2)) * TT + t) * VV;
      float psum = 0.f, psum2 = 0.f;
      if (mi2 == 0) {                             // uniform: i = g+8*lh in [0,16) always valid
        #pragma unroll
        for (int g = 0; g < 8; ++g) {
          float val = acc[g];
          ws[ybase + (g + 8 * lh)] = val;
          psum += val; psum2 += val * val;
        }
      } else {                                    // i = 16+g+8*lh, valid while < 25
        #pragma unroll
        for (int g = 0; g < 8; ++g) {
          int i = 16 + g + 8 * lh;
          if (i < VV) {
            float val = acc[g];
            ws[ybase + i] = val;
            psum += val; psum2 += val * val;
          }
        }
      }
      atomicAdd(&sY[o2], psum);
      atomicAdd(&sY2[o2], psum2);
    }
  }

  __syncthreads();
  for (int o = tid; o < CO; o += 256) {           // one global atomic per channel per WG
    atomicAdd(&ws[o],        sY[o]);
    atomicAdd(&ws[128 + o],  sY2[o]);
    atomicAdd(&ws[256 + o],  sR[o]);
    atomicAdd(&ws[384 + o],  sR2[o]);
  }
}

// ---------------------------------------------------------------- K2: BN coefficients
__global__ void k2_coef(const float* __restrict__ gamma,   const float* __restrict__ beta,
                        const float* __restrict__ gamma_r, const float* __restrict__ beta_r,
                        float* __restrict__ ws)
{
  int o = threadIdx.x;
  if (o < CO) {
    const float inv_cnt = 1.0f / ((float)R * (float)VV);   // 409600 elems per channel
    float my = ws[o]       * inv_cnt;
    float vy = ws[128 + o] * inv_cnt - my * my;
    float ay = gamma[o] * rsqrtf(vy + EPSV);
    float by = beta[o] - ay * my;
    float mr = ws[256 + o] * inv_cnt;
    float vr = ws[384 + o] * inv_cnt - mr * mr;
    float ar = gamma_r[o] * rsqrtf(vr + EPSV);
    float b2 = beta_r[o] - ar * mr;
    ws[COEF_OFF + o]       = ay;
    ws[COEF_OFF + 128 + o] = by;
    ws[COEF_OFF + 256 + o] = ar;
    ws[COEF_OFF + 384 + o] = b2;
  }
}

// ---------------------------------------------------------------- K3: BN(y)+BN(r)+ReLU, float4
__global__ __launch_bounds__(256)
void k3_bn_relu(const float* __restrict__ ws, float* __restrict__ out)
{
  size_t idx4 = (size_t)blockIdx.x * 256 + threadIdx.x;
  size_t idx  = idx4 * 4;
  int o = (int)((idx / (TT * VV)) % CO);          // TT*VV=1600 divisible by 4 -> uniform per vec
  float ay = ws[COEF_OFF + o],       by = ws[COEF_OFF + 128 + o];
  float ar = ws[COEF_OFF + 256 + o], b2 = ws[COEF_OFF + 384 + o];
  const float4* y4 = (const float4*)(ws + Y_OFF);
  const float4* r4 = (const float4*)(ws + R_OFF);
  float4 y = y4[idx4], r = r4[idx4], res;
  res.x = fmaxf(ay * y.x + by + ar * r.x + b2, 0.f);
  res.y = fmaxf(ay * y.y + by + ar * r.y + b2, 0.f);
  res.z = fmaxf(ay * y.z + by + ar * r.z + b2, 0.f);
  res.w = fmaxf(ay * y.w + by + ar * r.w + b2, 0.f);
  ((float4*)out)[idx4] = res;
}

// ----------------------------------------------------------------
extern "C" void kernel_launch(void* const* d_in, const int* in_sizes, int n_in,
                              void* d_out, int out_size, void* d_ws, size_t ws_size,
                              hipStream_t stream) {
  const float* x       = (const float*)d_in[0];
  const float* PA      = (const float*)d_in[1];
  const float* Wv      = (const float*)d_in[2];
  const float* bv      = (const float*)d_in[3];
  const float* gamma   = (const float*)d_in[4];
  const float* beta    = (const float*)d_in[5];
  const float* Wr      = (const float*)d_in[6];
  const float* br      = (const float*)d_in[7];
  const float* gamma_r = (const float*)d_in[8];
  const float* beta_r  = (const float*)d_in[9];
  float* ws  = (float*)d_ws;
  float* out = (float*)d_out;

  k0_zero_stats<<<4, 256, 0, stream>>>(ws);
  gsa_fused_k1<<<R / RB, 256, 0, stream>>>(x, PA, Wv, bv, Wr, br, ws);
  k2_coef<<<1, 128, 0, stream>>>(gamma, beta, gamma_r, beta_r, ws);
  k3_bn_relu<<<(int)(Y_ELEMS / 4 / 256), 256, 0, stream>>>(ws, out);
}